// Quantizer_84799834293024
// MI455X (gfx1250) — compile-verified
//
#include <hip/hip_runtime.h>
#include <hip/hip_bf16.h>

#define DIM    256
#define NEMB   8192
#define NROWS  32768
#define MTILE  128         // rows per workgroup (8 waves x 16 rows)
#define NCHUNK 32          // codebook columns per iteration (2 subtiles per wave)
#define LDA    260         // padded A stride (floats): 260 mod 64 = 4 -> conflict-free
#define LDBP   260         // padded B panel row stride [j][k] (TDM pad: 4 DW per 256 DW)

#define PANEL_FLOATS (NCHUNK * LDBP)                   // 8320 floats = 33280 B
#define SMEM_FLOATS  (MTILE * LDA + 2 * PANEL_FLOATS)  // 133120 B + 66560 B
#define SMEM_BYTES   (SMEM_FLOATS * 4)

typedef __attribute__((ext_vector_type(2))) float v2f;
typedef __attribute__((ext_vector_type(4))) float v4f;
typedef __attribute__((ext_vector_type(8))) float v8f;
typedef __attribute__((ext_vector_type(4))) unsigned int v4u;
typedef __attribute__((ext_vector_type(4))) int v4i;
typedef __attribute__((ext_vector_type(8))) int v8i;

#if defined(__HIP_DEVICE_COMPILE__) && defined(__has_builtin)
#  if __has_builtin(__builtin_amdgcn_tensor_load_to_lds) && \
      __has_builtin(__builtin_amdgcn_s_wait_tensorcnt)
#    define HAVE_TDM 1
#  else
#    define HAVE_TDM 0
#  endif
#else
#  define HAVE_TDM 0
#endif

// ---------------------------------------------------------------------------
// Phase 1a: e_sq[k] = sum_d embed[d,k]^2   (coalesced along k)
// ---------------------------------------------------------------------------
__global__ __launch_bounds__(256) void esq_kernel(const float* __restrict__ embed,
                                                  float* __restrict__ esq) {
    int k = blockIdx.x * 256 + threadIdx.x;
    float acc = 0.0f;
    #pragma unroll 8
    for (int d = 0; d < DIM; ++d) {
        float v = embed[d * NEMB + k];
        acc = fmaf(v, v, acc);
    }
    esq[k] = acc;
}

// ---------------------------------------------------------------------------
// Phase 1b: embed_T[k,d] = embed[d,k]  (LDS-tiled transpose, both sides coalesced)
// embed_T feeds BOTH the WMMA B-panels (contiguous row tiles) and the final gather.
// ---------------------------------------------------------------------------
__global__ __launch_bounds__(256) void transpose_kernel(const float* __restrict__ embed,
                                                        float* __restrict__ embed_T) {
    __shared__ float tile[32][33];
    int k0 = blockIdx.x * 32;
    int d0 = blockIdx.y * 32;
    int tx = threadIdx.x;          // 0..31
    int ty = threadIdx.y;          // 0..7
    #pragma unroll
    for (int i = 0; i < 32; i += 8)
        tile[ty + i][tx] = embed[(d0 + ty + i) * NEMB + (k0 + tx)];
    __syncthreads();
    #pragma unroll
    for (int i = 0; i < 32; i += 8)
        embed_T[(k0 + ty + i) * DIM + (d0 + tx)] = tile[tx][ty + i];
}

#if HAVE_TDM
// ---------------------------------------------------------------------------
// TDM DMA: 32 contiguous rows (256 f32 each) of embed_T -> LDS, with hardware
// padding of 4 DWORDs every 256 DWORDs => LDS row stride 260 floats, which is
// bank-conflict-free for the WMMA fragment reads.
// D# per CDNA5 ISA 8.3/8.4: count=1, data_size=4B, type=2,
// tensor 256x32 (stride 256), tile 256x32, pad_interval=7 (256DW), pad_amount=3 (4DW).
// ---------------------------------------------------------------------------
__device__ __forceinline__ void tdm_load_panel(unsigned lds_byte_off,
                                               const float* gaddr) {
    unsigned long long ga = (unsigned long long)(size_t)gaddr;
    v4u g0;
    g0[0] = 1u;                                           // count=1 (valid), user mode
    g0[1] = lds_byte_off;                                 // lds_addr
    g0[2] = (unsigned)(ga & 0xFFFFFFFFu);                 // global_addr[31:0]
    g0[3] = (unsigned)((ga >> 32) & 0x01FFFFFFu)          // global_addr[56:32]
          | (2u << 30);                                   // type = 2
    v8i g1;
    g1[0] = (int)((2u << 16)          // data_size = 2 (4 bytes)
                | (1u << 20)          // pad_enable
                | (7u << 22)          // pad_interval: 256 DWORDs
                | (3u << 25));        // pad_amount:   4 DWORDs
    g1[1] = (int)(256u << 16);        // tensor_dim0 = 256 (low16 in [31:16])
    g1[2] = (int)(32u << 16);         // tensor_dim0 hi=0; tensor_dim1 = 32 (low16)
    g1[3] = (int)(256u << 16);        // tensor_dim1 hi=0; tile_dim0 = 256
    g1[4] = (int)32;                  // tile_dim1 = 32; tile_dim2 = 0
    g1[5] = (int)256;                 // tensor_dim0_stride = 256 (low 32)
    g1[6] = 0;
    g1[7] = 0;
    v4i z = {0, 0, 0, 0};
#if defined(__clang_major__) && (__clang_major__ >= 23)
    v8i z8 = {0, 0, 0, 0, 0, 0, 0, 0};
    __builtin_amdgcn_tensor_load_to_lds(g0, g1, z, z, z8, 0);
#else
    __builtin_amdgcn_tensor_load_to_lds(g0, g1, z, z, 0);
#endif
}
#endif // HAVE_TDM

// ---------------------------------------------------------------------------
// Phase 2: fused GEMM + online argmin.
//   score[n,k] = e_sq[k] - 2 * dot(x_n, e_k)   (x^2 constant per row: dropped)
// 8 waves/WG, 128-row A tile in LDS. K swept in 32-col panels from embed_T,
// double-buffered: wave 0 issues tensor_load_to_lds (with pad-swizzle) for
// panel p^1 while all waves run 128 V_WMMA_F32_16X16X4_F32 on panel p.
// A/B0/B1 each use an opaque loop-invariant LDS base (asm barrier) so every
// fragment load folds its offset into the ds_load_2addr immediate fields.
// ---------------------------------------------------------------------------
__global__ __launch_bounds__(256) void vq_argmin_kernel(const float* __restrict__ x,
                                                        const float* __restrict__ embed_T,
                                                        const float* __restrict__ esq,
                                                        int* __restrict__ ind) {
    extern __shared__ float smem[];
    float* Alds = smem;                               // MTILE x LDA
    float* Bt   = smem + MTILE * LDA;                 // 2 panels, each [j][k] 32x260
    const unsigned BtOffBytes = (unsigned)(MTILE * LDA * 4);

    const int t    = threadIdx.x;
    const int wave = t >> 5;
    const int lane = t & 31;
    const int l16  = lane & 15;
    const int hi   = lane >> 4;                 // 0: lanes 0-15, 1: lanes 16-31
    const int rowbase = blockIdx.x * MTILE;
    const int mwave   = wave * 16;

#if HAVE_TDM
    if (wave == 0)
        tdm_load_panel(BtOffBytes, embed_T);          // prefetch panel for n0 = 0
#endif

    // Stage A tile (128 rows x 256 cols) as float4, coalesced, padded stride.
    for (int e = t; e < MTILE * (DIM / 4); e += 256) {
        int r  = e >> 6;
        int c4 = e & 63;
        v4f v = *(const v4f*)&x[(rowbase + r) * DIM + c4 * 4];
        *(v4f*)&Alds[r * LDA + c4 * 4] = v;
    }

#if HAVE_TDM
    if (wave == 0)
        __builtin_amdgcn_s_wait_tensorcnt(0);
#endif
    __syncthreads();

    float best[8];
    int   bidx[8];
    #pragma unroll
    for (int v = 0; v < 8; ++v) { best[v] = 3.4e38f; bidx[v] = 0; }

    // A fragment: lane holds row (mwave+l16); VGPR0 = K {0,2}, VGPR1 = K {1,3}
    unsigned offA = (unsigned)((mwave + l16) * LDA + 2 * hi);
    asm volatile("" : "+v"(offA));                    // pin base in a VGPR
    const float* Abase = &Alds[offA];

    for (int n0 = 0; n0 < NEMB; n0 += NCHUNK) {
        const int cur = (n0 >> 5) & 1;
        float* Bp = Bt + cur * PANEL_FLOATS;

#if HAVE_TDM
        // Kick DMA for the next panel into the other buffer while we compute.
        if (wave == 0 && (n0 + NCHUNK) < NEMB)
            tdm_load_panel(BtOffBytes + (unsigned)((cur ^ 1) * PANEL_FLOATS * 4),
                           embed_T + (size_t)(n0 + NCHUNK) * DIM);
#else
        // Fallback: stage padded [j][k] panel from embed_T (reads coalesced).
        __syncthreads();
        for (int e = t; e < NCHUNK * DIM; e += 256) {
            int j = e >> 8;
            int c = e & 255;
            Bp[j * LDBP + c] = embed_T[(size_t)(n0 + j) * DIM + c];
        }
        __syncthreads();
#endif

        // B fragments: [j][k] panel, stride LDBP. b.x/b.y adjacent floats.
        // Opaque, independent bases -> no reassociation, immediate ds offsets.
        unsigned offB0 = (unsigned)(cur * PANEL_FLOATS + l16 * LDBP + 2 * hi);
        unsigned offB1 = offB0 + 16u * LDBP;
        asm volatile("" : "+v"(offB0));
        asm volatile("" : "+v"(offB1));
        const float* B0 = &Bt[offB0];                    // col n0+l16
        const float* B1 = &Bt[offB1];                    // col n0+16+l16

        v8f acc0 = {0.f, 0.f, 0.f, 0.f, 0.f, 0.f, 0.f, 0.f};
        v8f acc1 = {0.f, 0.f, 0.f, 0.f, 0.f, 0.f, 0.f, 0.f};
        #pragma unroll
        for (int kk = 0; kk < DIM / 4; ++kk) {           // fully unrolled: 4kk <= 252
            v2f a, b0, b1;
            a.x  = Abase[kk * 4 + 0];         // K = 4kk + 2*hi
            a.y  = Abase[kk * 4 + 1];         // K = 4kk + 2*hi + 1
            b0.x = B0[kk * 4 + 0];
            b0.y = B0[kk * 4 + 1];
            b1.x = B1[kk * 4 + 0];
            b1.y = B1[kk * 4 + 1];
            acc0 = __builtin_amdgcn_wmma_f32_16x16x4_f32(
                       false, a, false, b0, (short)0, acc0, false, false);
            acc1 = __builtin_amdgcn_wmma_f32_16x16x4_f32(
                       false, a, false, b1, (short)0, acc1, false, false);
        }

        // C layout: VGPR v, lanes 0-15 -> (M=v, N=l16); lanes 16-31 -> (M=v+8, N=l16)
        float esq0 = esq[n0 + l16];
        float esq1 = esq[n0 + 16 + l16];
        #pragma unroll
        for (int v = 0; v < 8; ++v) {
            float s0 = fmaf(-2.0f, acc0[v], esq0);
            if (s0 < best[v]) { best[v] = s0; bidx[v] = n0 + l16; }
            float s1 = fmaf(-2.0f, acc1[v], esq1);
            if (s1 < best[v]) { best[v] = s1; bidx[v] = n0 + 16 + l16; }
        }

#if HAVE_TDM
        // Ensure next panel's DMA has landed before anyone reads it.
        if (wave == 0)
            __builtin_amdgcn_s_wait_tensorcnt(0);
        __syncthreads();
#endif
    }

    // Min+index reduction across the 16 lanes sharing each row (wave32 halves).
    #pragma unroll
    for (int v = 0; v < 8; ++v) {
        float b_ = best[v];
        int   i_ = bidx[v];
        #pragma unroll
        for (int m = 8; m >= 1; m >>= 1) {
            float ob = __shfl_xor(b_, m, 32);
            int   oi = __shfl_xor(i_, m, 32);
            if (ob < b_ || (ob == b_ && oi < i_)) { b_ = ob; i_ = oi; }
        }
        if (l16 == 0)
            ind[rowbase + mwave + hi * 8 + v] = i_;
    }
}

// ---------------------------------------------------------------------------
// Phase 3: coalesced gather of winning codes: out[n, :] = embed_T[ind[n], :]
// ---------------------------------------------------------------------------
__global__ __launch_bounds__(256) void gather_kernel(const int* __restrict__ ind,
                                                     const float* __restrict__ embed_T,
                                                     float* __restrict__ out) {
    int n = blockIdx.x;
    int k = ind[n];
    out[n * DIM + threadIdx.x] = embed_T[(size_t)k * DIM + threadIdx.x];
}

// ---------------------------------------------------------------------------
extern "C" void kernel_launch(void* const* d_in, const int* in_sizes, int n_in,
                              void* d_out, int out_size, void* d_ws, size_t ws_size,
                              hipStream_t stream) {
    const float* x     = (const float*)d_in[0];   // [32,32,32,256] f32
    const float* embed = (const float*)d_in[1];   // [256, 8192]   f32
    float* out = (float*)d_out;

    // Workspace layout: esq[8192] f32 | ind[32768] i32 | embed_T[8192*256] f32
    float* esq     = (float*)d_ws;
    int*   ind     = (int*)((char*)d_ws + 32768);
    float* embed_T = (float*)((char*)d_ws + 163840);

    esq_kernel<<<NEMB / 256, 256, 0, stream>>>(embed, esq);

    dim3 tgrid(NEMB / 32, DIM / 32);
    dim3 tblk(32, 8);
    transpose_kernel<<<tgrid, tblk, 0, stream>>>(embed, embed_T);

    vq_argmin_kernel<<<NROWS / MTILE, 256, SMEM_BYTES, stream>>>(x, embed_T, esq, ind);

    gather_kernel<<<NROWS, DIM, 0, stream>>>(ind, embed_T, out);
}